// SAGE_77464030151198
// MI455X (gfx1250) — compile-verified
//
#include <hip/hip_runtime.h>
#include <hip/hip_bf16.h>

// ---------------------------------------------------------------------------
// GraphSAGE (3 layers, mean aggregator) for MI455X / gfx1250.
// N=50000 nodes, 800000 edges, feature dims 128 -> 128 -> 128 -> 47.
//
// Memory-bound workload: per layer ~410MB random gather + ~410MB f32 atomics
// (agg buffer = 25.6MB, L2-resident on the 192MB L2). GEMMs are exact-fp32
// via V_WMMA_F32_16X16X4_F32 (one wave per 16x16 output tile, K=2*128),
// fully unrolled: 64 straight-line WMMAs with immediate-offset b64 loads.
// ---------------------------------------------------------------------------

#define N_NODES 50000
#define N_EDGES 800000
#define D_IN    128
#define D_HID   128
#define D_OUT   47

typedef __attribute__((ext_vector_type(2))) float v2f;
typedef __attribute__((ext_vector_type(8))) float v8f;

// ----------------------------- utility kernels -----------------------------

__global__ void sage_zero_f32(float* __restrict__ p, int n) {
    int i = blockIdx.x * blockDim.x + threadIdx.x;
    if (i < n) p[i] = 0.0f;
}

__global__ void sage_count_deg(const int* __restrict__ dst,
                               float* __restrict__ deg, int n_edges) {
    int e = blockIdx.x * blockDim.x + threadIdx.x;
    if (e < n_edges) atomicAdd(deg + dst[e], 1.0f);
}

__global__ void sage_recip_deg(float* __restrict__ deg, int n) {
    int i = blockIdx.x * blockDim.x + threadIdx.x;
    if (i < n) deg[i] = 1.0f / fmaxf(deg[i], 1.0f);
}

// One wave per edge: 32 lanes x float4 = 128 features.
__global__ void sage_scatter_add(const float* __restrict__ h,
                                 const int* __restrict__ src,
                                 const int* __restrict__ dst,
                                 float* __restrict__ agg, int n_edges) {
    int wave_in_blk = threadIdx.x >> 5;
    int lane        = threadIdx.x & 31;
    int e = blockIdx.x * (blockDim.x >> 5) + wave_in_blk;
    if (e >= n_edges) return;
    int s = src[e];
    int t = dst[e];
    const float4 v = *(const float4*)(h + (size_t)s * D_HID + lane * 4);
    float* p = agg + (size_t)t * D_HID + lane * 4;
    atomicAdd(p + 0, v.x);
    atomicAdd(p + 1, v.y);
    atomicAdd(p + 2, v.z);
    atomicAdd(p + 3, v.w);
}

// agg[n, :] *= invdeg[n]   (in place -> mean)
__global__ void sage_scale_rows(float* __restrict__ agg,
                                const float* __restrict__ invdeg, int total) {
    int i = blockIdx.x * blockDim.x + threadIdx.x;
    if (i < total) agg[i] *= invdeg[i >> 7];   // 128 feats per row
}

// ------------------------------- WMMA GEMM ---------------------------------
// out[n, o] = relu?( h[n,:]·w_self[o,:] + mean[n,:]·w_neigh[o,:] + b[o] )
// One wave (32 threads) per 16x16 tile of out. Two fully-unrolled K=128
// halves (uniform pointers, no per-iteration selects), 32 K-steps each,
// 64 v_wmma_f32_16x16x4_f32 total. Exact fp32 = reference arithmetic.
//
// A fragment (16x4 f32, ISA 7.12.2): lane l<16 -> row M=l, holds K=0,1 in
// v[0],v[1]; lanes 16-31 -> row M=l-16, holds K=2,3. B fragment mirrored
// (lane = output column, K halves split by lane group). C/D: element (M,N)
// lives in lane N + 16*(M/8), vgpr M%8.
//
// Column masking: B column n only feeds output column n (lanes n, n+16), and
// columns >= d_out are never stored, so we simply CLAMP the B row index and
// load unconditionally -- no EXEC-mask branches in the hot loop.
__global__ void sage_wmma_gemm(const float* __restrict__ h,
                               const float* __restrict__ mean,
                               const float* __restrict__ w_self,
                               const float* __restrict__ w_neigh,
                               const float* __restrict__ bias,
                               float* __restrict__ out,
                               int d_out, int do_relu) {
    const int lane = threadIdx.x & 31;
    const int half = lane >> 4;                 // 0: lanes 0-15, 1: lanes 16-31
    const int sub  = lane & 15;
    const int row  = blockIdx.x * 16 + sub;     // A row (node index)
    const int col  = blockIdx.y * 16 + sub;     // B col (output feature)
    const int kh   = half * 2;                  // per-lane K sub-offset
    const int colc = (col < d_out) ? col : (d_out - 1);   // clamped, in-bounds

    v8f acc = {};

    // Loop-invariant per-lane base pointers -> immediate-offset loads.
    const float* __restrict__ pa0 = h       + (size_t)row  * D_HID + kh;
    const float* __restrict__ pb0 = w_self  + (size_t)colc * D_HID + kh;
    const float* __restrict__ pa1 = mean    + (size_t)row  * D_HID + kh;
    const float* __restrict__ pb1 = w_neigh + (size_t)colc * D_HID + kh;

    #pragma unroll
    for (int k0 = 0; k0 < D_HID; k0 += 4) {
        v2f a = *(const v2f*)(pa0 + k0);
        v2f b = *(const v2f*)(pb0 + k0);
        acc = __builtin_amdgcn_wmma_f32_16x16x4_f32(
            false, a, false, b, (short)0, acc, false, false);
    }
    #pragma unroll
    for (int k0 = 0; k0 < D_HID; k0 += 4) {
        v2f a = *(const v2f*)(pa1 + k0);
        v2f b = *(const v2f*)(pb1 + k0);
        acc = __builtin_amdgcn_wmma_f32_16x16x4_f32(
            false, a, false, b, (short)0, acc, false, false);
    }

    // Store D: lane holds column n = sub, rows m = v + 8*half.
    const int ocol = blockIdx.y * 16 + sub;
    if (ocol >= d_out) return;                   // drop garbage columns
    const float bval = bias[ocol];
    #pragma unroll
    for (int v = 0; v < 8; ++v) {
        const int m    = v + 8 * half;
        const int orow = blockIdx.x * 16 + m;
        float val = acc[v] + bval;
        if (do_relu) val = fmaxf(val, 0.0f);
        out[(size_t)orow * d_out + ocol] = val;
    }
}

// ------------------------------- launcher ----------------------------------

extern "C" void kernel_launch(void* const* d_in, const int* in_sizes, int n_in,
                              void* d_out, int out_size, void* d_ws, size_t ws_size,
                              hipStream_t stream) {
    const float* x        = (const float*)d_in[0];
    const int*   src      = (const int*)  d_in[1];
    const int*   dst      = (const int*)  d_in[2];
    const float* w_self1  = (const float*)d_in[3];
    const float* w_neigh1 = (const float*)d_in[4];
    const float* b1       = (const float*)d_in[5];
    const float* w_self2  = (const float*)d_in[6];
    const float* w_neigh2 = (const float*)d_in[7];
    const float* b2       = (const float*)d_in[8];
    const float* w_self3  = (const float*)d_in[9];
    const float* w_neigh3 = (const float*)d_in[10];
    const float* b3       = (const float*)d_in[11];
    float* out = (float*)d_out;

    // Workspace layout (floats): h1 | h2 | agg | invdeg  ~= 77 MB
    float* h1     = (float*)d_ws;
    float* h2     = h1  + (size_t)N_NODES * D_HID;
    float* agg    = h2  + (size_t)N_NODES * D_HID;
    float* invdeg = agg + (size_t)N_NODES * D_HID;

    const int feat_total = N_NODES * D_HID;            // 6.4M
    const int TPB = 256;
    const int waves_per_blk = TPB / 32;
    dim3 edge_grid((N_EDGES + waves_per_blk - 1) / waves_per_blk);
    dim3 gemm_grid_hid(N_NODES / 16, D_HID / 16);        // 3125 x 8
    dim3 gemm_grid_out(N_NODES / 16, (D_OUT + 15) / 16); // 3125 x 3

    // --- degrees (shared by all layers) ---
    sage_zero_f32<<<(N_NODES + TPB - 1) / TPB, TPB, 0, stream>>>(invdeg, N_NODES);
    sage_count_deg<<<(N_EDGES + TPB - 1) / TPB, TPB, 0, stream>>>(dst, invdeg, N_EDGES);
    sage_recip_deg<<<(N_NODES + TPB - 1) / TPB, TPB, 0, stream>>>(invdeg, N_NODES);

    // --- layer 1: x -> h1 (relu) ---
    sage_zero_f32<<<(feat_total + TPB - 1) / TPB, TPB, 0, stream>>>(agg, feat_total);
    sage_scatter_add<<<edge_grid, TPB, 0, stream>>>(x, src, dst, agg, N_EDGES);
    sage_scale_rows<<<(feat_total + TPB - 1) / TPB, TPB, 0, stream>>>(agg, invdeg, feat_total);
    sage_wmma_gemm<<<gemm_grid_hid, 32, 0, stream>>>(x, agg, w_self1, w_neigh1, b1,
                                                     h1, D_HID, /*relu=*/1);

    // --- layer 2: h1 -> h2 (relu) ---
    sage_zero_f32<<<(feat_total + TPB - 1) / TPB, TPB, 0, stream>>>(agg, feat_total);
    sage_scatter_add<<<edge_grid, TPB, 0, stream>>>(h1, src, dst, agg, N_EDGES);
    sage_scale_rows<<<(feat_total + TPB - 1) / TPB, TPB, 0, stream>>>(agg, invdeg, feat_total);
    sage_wmma_gemm<<<gemm_grid_hid, 32, 0, stream>>>(h1, agg, w_self2, w_neigh2, b2,
                                                     h2, D_HID, /*relu=*/1);

    // --- layer 3: h2 -> out (no relu, d_out = 47) ---
    sage_zero_f32<<<(feat_total + TPB - 1) / TPB, TPB, 0, stream>>>(agg, feat_total);
    sage_scatter_add<<<edge_grid, TPB, 0, stream>>>(h2, src, dst, agg, N_EDGES);
    sage_scale_rows<<<(feat_total + TPB - 1) / TPB, TPB, 0, stream>>>(agg, invdeg, feat_total);
    sage_wmma_gemm<<<gemm_grid_out, 32, 0, stream>>>(h2, agg, w_self3, w_neigh3, b3,
                                                     out, D_OUT, /*relu=*/0);
    (void)in_sizes; (void)n_in; (void)out_size; (void)ws_size;
}